// NECDGT_batching_81088982548656
// MI455X (gfx1250) — compile-verified
//
#include <hip/hip_runtime.h>
#include <math.h>

// ---------------------------------------------------------------------------
// NECDGT graph-network forward for MI455X (gfx1250, wave32).
//
// Strategy (see analysis): R/S are one-hot incidence matrices -> recover
// recv/send indices with one streaming scan (512MB @ 23.3TB/s ~ 22us), then
// run every MLP as wave-level fp32 WMMA (V_WMMA_F32_16X16X4_F32) on 16-row
// tiles staged in LDS; edge->node reductions via global_atomic_add_f32;
// softmax over the edge dim with one block per (batch, feature).
// ---------------------------------------------------------------------------

typedef __attribute__((ext_vector_type(2))) float v2f;
typedef __attribute__((ext_vector_type(8))) float v8f;

constexpr int kN = 4096, kM = 16384, kB = 8;
constexpr int kFV = 16, kFE = 8, kQ = 20, kINTER = 5;

constexpr int WPB = 4;           // waves per block (wave32)
constexpr int TPB = WPB * 32;

__host__ __device__ constexpr int roundup(int x, int a) { return (x + a - 1) & ~(a - 1); }
// LDS floats needed for a padded layer: W^T padded to [din4][dout16] + bias[dout16]
__host__ __device__ constexpr int lsz(int din, int dout) {
  return roundup(din, 4) * roundup(dout, 16) + roundup(dout, 16);
}
constexpr int WSLOT = lsz(50, 50);   // 3392 floats: largest layer (50->50)

// ---- stage W (row-major [dout][din]) as zero-padded W^T [din4][dout16] + bias
__device__ __forceinline__ void stage_lin(float* dst, const float* __restrict__ W,
                                          const float* __restrict__ bias,
                                          int din, int dout, int tid) {
  const int din4 = roundup(din, 4), dout16 = roundup(dout, 16);
  const int total = din4 * dout16;
  for (int i = tid; i < total; i += TPB) {
    const int k = i / dout16, n = i - k * dout16;
    dst[i] = (k < din && n < dout) ? W[n * din + k] : 0.0f;
  }
  for (int n = tid; n < dout16; n += TPB)
    dst[total + n] = (n < dout) ? bias[n] : 0.0f;
}

// shared-weight concat layer: W = [Wrep | Wrep | ... | Wtail] along K
__device__ __forceinline__ void stage_cat2(float* dst, const float* __restrict__ Wrep,
                                           int repc, int nrep,
                                           const float* __restrict__ Wtail, int tailc,
                                           const float* __restrict__ bias, int dout, int tid) {
  const int din = repc * nrep + tailc;
  const int din4 = roundup(din, 4), dout16 = roundup(dout, 16);
  const int total = din4 * dout16;
  for (int i = tid; i < total; i += TPB) {
    const int k = i / dout16, n = i - k * dout16;
    float v = 0.0f;
    if (k < din && n < dout)
      v = (k < repc * nrep) ? Wrep[n * repc + (k % repc)]
                            : Wtail[n * tailc + (k - repc * nrep)];
    dst[i] = v;
  }
  for (int n = tid; n < dout16; n += TPB)
    dst[total + n] = (n < dout) ? bias[n] : 0.0f;
}

// ---- wave-level 16-row GEMM: xout[16][dout] = act(xin[16][din] @ Wpad + bias)
// xin/xout: per-wave LDS tiles, row stride 64 floats; wpad: staged weights.
// Uses V_WMMA_F32_16X16X4_F32. Fragment layouts per CDNA5 ISA 7.12.2:
//   A: lane l -> row m=l&15, a[r] = X[m][k0 + 2*(l>>4) + r]
//   B: lane l -> col n=l&15, b[r] = Wt[k0 + 2*(l>>4) + r][n]
//   C/D: VGPR r -> row r + 8*(l>>4), col l&15 (per 16-col tile)
__device__ __forceinline__ void wave_gemm(const float* __restrict__ wpad,
                                          const float* __restrict__ xin,
                                          float* __restrict__ xout,
                                          int din4, int dout, bool relu, int lane) {
  const int dout16 = roundup(dout, 16);
  const float* bias = wpad + din4 * dout16;
  const int half = lane >> 4;
  const int l15  = lane & 15;
  for (int nt = 0; nt < dout16; nt += 16) {
    v8f acc = {};
    const int n = nt + l15;
    for (int k0 = 0; k0 < din4; k0 += 4) {
      const int ka = k0 + 2 * half;
      v2f a, b;
      a.x = xin[l15 * 64 + ka];
      a.y = xin[l15 * 64 + ka + 1];
      b.x = wpad[ka * dout16 + n];
      b.y = wpad[(ka + 1) * dout16 + n];
      acc = __builtin_amdgcn_wmma_f32_16x16x4_f32(false, a, false, b,
                                                  (short)0, acc, false, false);
    }
    const float bv = bias[n];
#pragma unroll
    for (int r = 0; r < 8; ++r) {
      float v = acc[r] + bv;
      if (relu) v = fmaxf(v, 0.0f);
      xout[(r + 8 * half) * 64 + n] = v;   // cols [dout,dout16) get exact zeros
    }
  }
  // zero-pad columns [dout16, 64) so the next layer's padded-K reads see zeros
  for (int c = dout16 + lane; c < 64; c += 32)
    for (int mr = 0; mr < 16; ++mr) xout[mr * 64 + c] = 0.0f;
}

// stage weights (block-wide) + run per-wave GEMM. Uniform across all waves.
__device__ __forceinline__ void layer(float* wslot, const float* W, const float* b,
                                      int din, int dout, const float* xin, float* xout,
                                      bool relu, int tid, int lane) {
  __syncthreads();                       // previous users of wslot are done
  stage_lin(wslot, W, b, din, dout, tid);
  __syncthreads();
  wave_gemm(wslot, xin, xout, roundup(din, 4), dout, relu, lane);
}

// ---------------------------------------------------------------------------
// Kernel 0: recover recv/send indices from one-hot incidence columns.
// ---------------------------------------------------------------------------
__global__ __launch_bounds__(256) void k_extract_idx(const float* __restrict__ R,
                                                     const float* __restrict__ S,
                                                     int* __restrict__ recv,
                                                     int* __restrict__ send) {
  const int m = blockIdx.x * blockDim.x + threadIdx.x;
  if (m >= kM) return;
  int r = 0, s = 0;
  for (int n = 0; n < kN; ++n) {         // coalesced: lanes read consecutive m
    if (R[(size_t)n * kM + m] != 0.0f) r = n;
    if (S[(size_t)n * kM + m] != 0.0f) s = n;
  }
  recv[m] = r;
  send[m] = s;
}

// ---------------------------------------------------------------------------
// Kernel 1: edge stage 1.  Per 16-edge tile:
//   B1t=[X[recv],X[send],Ra]; EO1=ve1(B1t) -> atomic scatter into Hev;
//   He=relu(B1t@W1^T+b1); ER1=ee1(He); He2=ee2([ER1,Ra]) -> ws.
// ---------------------------------------------------------------------------
struct S1P {
  const float *node, *Ra;
  const int *recv, *send;
  const float *ve10W, *ve10b, *ve11W, *ve11b, *ve12W, *ve12b;
  const float *w11, *w12, *b1;
  const float *ee10W, *ee10b, *ee11W, *ee11b, *ee20W, *ee20b, *ee21W, *ee21b;
  float *Hev, *He2;
};

__global__ __launch_bounds__(TPB) void k_edge_stage1(S1P p) {
  __shared__ float wslot[WSLOT];
  __shared__ float bufs[WPB][3][16 * 64];
  const int tid = threadIdx.x, lane = tid & 31, wave = tid >> 5;
  float* A  = &bufs[wave][0][0];
  float* Bu = &bufs[wave][1][0];
  float* C  = &bufs[wave][2][0];
  const int row0 = (blockIdx.x * WPB + wave) * 16;   // rows over B*M

  // build B1t (cols 0-15 XR | 16-31 XS | 32-39 Ra); cols>=40 never read (din4=40)
  for (int i = lane; i < 16 * 40; i += 32) {
    const int r = i / 40, c = i - r * 40;
    const int row = row0 + r, b = row >> 14, m = row & (kM - 1);
    float v;
    if (c < 16)      v = p.node[((size_t)b * kN + p.recv[m]) * kFV + c];
    else if (c < 32) v = p.node[((size_t)b * kN + p.send[m]) * kFV + (c - 16)];
    else             v = p.Ra[(size_t)row * kFE + (c - 32)];
    A[r * 64 + c] = v;
  }

  layer(wslot, p.ve10W, p.ve10b, 40, 50, A, Bu, true, tid, lane);
  layer(wslot, p.ve11W, p.ve11b, 50, 50, Bu, C, true, tid, lane);
  layer(wslot, p.ve12W, p.ve12b, 50, 20, C, Bu, false, tid, lane);  // EO1
  for (int i = lane; i < 16 * kQ; i += 32) {                        // scatter -> Hev
    const int r = i / kQ, q = i - r * kQ;
    const int row = row0 + r, b = row >> 14, m = row & (kM - 1);
    atomicAdd(&p.Hev[((size_t)b * kN + p.recv[m]) * kQ + q], Bu[r * 64 + q]);
  }

  // He = relu(B1t @ W1^T + b1), W1 = [w1_1 | w1_1 | w1_2]
  __syncthreads();
  stage_cat2(wslot, p.w11, 16, 2, p.w12, 8, p.b1, 50, tid);
  __syncthreads();
  wave_gemm(wslot, A, C, 40, 50, true, lane);

  layer(wslot, p.ee10W, p.ee10b, 50, 50, C, Bu, true, tid, lane);
  layer(wslot, p.ee11W, p.ee11b, 50, 20, Bu, C, false, tid, lane);  // ER1
  for (int i = lane; i < 16 * 28; i += 32) {                        // [ER1 | Ra]
    const int r = i / 28, c = i - r * 28;
    Bu[r * 64 + c] = (c < kQ) ? C[r * 64 + c] : A[r * 64 + 32 + (c - kQ)];
  }
  layer(wslot, p.ee20W, p.ee20b, 28, 50, Bu, C, true, tid, lane);
  layer(wslot, p.ee21W, p.ee21b, 50, 5, C, Bu, true, tid, lane);    // He2
  for (int i = lane; i < 16 * kINTER; i += 32) {
    const int r = i / kINTER, c = i - r * kINTER;
    p.He2[(size_t)(row0 + r) * kINTER + c] = Bu[r * 64 + c];
  }
}

// ---------------------------------------------------------------------------
// Kernels 2/4: node MLP  out = relu(L1(relu(L0([xa(20) | node(16)]))))
// (vv1: dout=5 -> Hv2, vv2: dout=16 -> Hv3)
// ---------------------------------------------------------------------------
__global__ __launch_bounds__(TPB) void k_node_mlp(const float* __restrict__ xa,
                                                  const float* __restrict__ node,
                                                  const float* l0W, const float* l0b,
                                                  const float* l1W, const float* l1b,
                                                  float* __restrict__ out, int dout) {
  __shared__ float wslot[WSLOT];
  __shared__ float bufs[WPB][2][16 * 64];
  const int tid = threadIdx.x, lane = tid & 31, wave = tid >> 5;
  float* A  = &bufs[wave][0][0];
  float* Bu = &bufs[wave][1][0];
  const int row0 = (blockIdx.x * WPB + wave) * 16;   // rows over B*N
  for (int i = lane; i < 16 * 36; i += 32) {
    const int r = i / 36, c = i - r * 36;
    const size_t row = (size_t)(row0 + r);
    A[r * 64 + c] = (c < kQ) ? xa[row * kQ + c] : node[row * kFV + (c - kQ)];
  }
  layer(wslot, l0W, l0b, 36, 50, A, Bu, true, tid, lane);
  layer(wslot, l1W, l1b, 50, dout, Bu, A, true, tid, lane);
  for (int i = lane; i < 16 * dout; i += 32) {
    const int r = i / dout, c = i - r * dout;
    out[(size_t)(row0 + r) * dout + c] = A[r * 64 + c];
  }
}

// ---------------------------------------------------------------------------
// Kernel 3: edge stage 2.
//   B2t=[Hv2[recv],Hv2[send],He2]; EO2=ve2(B2t)-> scatter Hev2;
//   He=relu(B2t@W2^T+b2); ER2=ee3(He); logits=ee4([ER2,Ra]) -> d_out.
// ---------------------------------------------------------------------------
struct S2P {
  const float *Hv2, *He2, *Ra;
  const int *recv, *send;
  const float *ve20W, *ve20b, *ve21W, *ve21b, *ve22W, *ve22b;
  const float *w21, *w22, *b2;
  const float *ee30W, *ee30b, *ee31W, *ee31b, *ee40W, *ee40b, *ee41W, *ee41b;
  float *Hev2, *logits;
};

__global__ __launch_bounds__(TPB) void k_edge_stage2(S2P p) {
  __shared__ float wslot[WSLOT];
  __shared__ float bufs[WPB][3][16 * 64];
  const int tid = threadIdx.x, lane = tid & 31, wave = tid >> 5;
  float* A  = &bufs[wave][0][0];
  float* Bu = &bufs[wave][1][0];
  float* C  = &bufs[wave][2][0];
  const int row0 = (blockIdx.x * WPB + wave) * 16;

  // A cols: 0-4 Hv2[recv] | 5-9 Hv2[send] | 10-14 He2 | 15 zero | 16-23 Ra stash
  for (int i = lane; i < 16 * 24; i += 32) {
    const int r = i / 24, c = i - r * 24;
    const int row = row0 + r, b = row >> 14, m = row & (kM - 1);
    float v;
    if (c < 5)       v = p.Hv2[((size_t)b * kN + p.recv[m]) * kINTER + c];
    else if (c < 10) v = p.Hv2[((size_t)b * kN + p.send[m]) * kINTER + (c - 5)];
    else if (c < 15) v = p.He2[(size_t)row * kINTER + (c - 10)];
    else if (c == 15) v = 0.0f;
    else             v = p.Ra[(size_t)row * kFE + (c - 16)];
    A[r * 64 + c] = v;
  }

  layer(wslot, p.ve20W, p.ve20b, 15, 50, A, Bu, true, tid, lane);
  layer(wslot, p.ve21W, p.ve21b, 50, 50, Bu, C, true, tid, lane);
  layer(wslot, p.ve22W, p.ve22b, 50, 20, C, Bu, false, tid, lane);  // EO2
  for (int i = lane; i < 16 * kQ; i += 32) {                        // scatter -> Hev2
    const int r = i / kQ, q = i - r * kQ;
    const int row = row0 + r, b = row >> 14, m = row & (kM - 1);
    atomicAdd(&p.Hev2[((size_t)b * kN + p.recv[m]) * kQ + q], Bu[r * 64 + q]);
  }

  __syncthreads();
  stage_cat2(wslot, p.w21, 5, 2, p.w22, 5, p.b2, 50, tid);          // W2=[w2_1|w2_1|w2_2]
  __syncthreads();
  wave_gemm(wslot, A, C, 16, 50, true, lane);                       // He

  layer(wslot, p.ee30W, p.ee30b, 50, 50, C, Bu, true, tid, lane);
  layer(wslot, p.ee31W, p.ee31b, 50, 20, Bu, C, false, tid, lane);  // ER2
  for (int i = lane; i < 16 * 28; i += 32) {                        // [ER2 | Ra]
    const int r = i / 28, c = i - r * 28;
    Bu[r * 64 + c] = (c < kQ) ? C[r * 64 + c] : A[r * 64 + 16 + (c - kQ)];
  }
  layer(wslot, p.ee40W, p.ee40b, 28, 50, Bu, C, true, tid, lane);
  layer(wslot, p.ee41W, p.ee41b, 50, 8, C, Bu, false, tid, lane);   // logits
  for (int i = lane; i < 16 * kFE; i += 32) {
    const int r = i / kFE, c = i - r * kFE;
    p.logits[(size_t)(row0 + r) * kFE + c] = Bu[r * 64 + c];
  }
}

// ---------------------------------------------------------------------------
// Kernel 5: softmax over the edge dimension (axis=1) per (batch, feature).
// ---------------------------------------------------------------------------
__global__ __launch_bounds__(256) void k_softmax_edges(const float* __restrict__ logits,
                                                       float* __restrict__ out) {
  const int b = blockIdx.x >> 3, f = blockIdx.x & 7;
  const float* base = logits + (size_t)b * kM * kFE + f;
  float* obase = out + (size_t)b * kM * kFE + f;
  __shared__ float red[256];
  const int tid = threadIdx.x;
  float mx = -3.4e38f;
  for (int m = tid; m < kM; m += 256) mx = fmaxf(mx, base[(size_t)m * kFE]);
  red[tid] = mx;
  __syncthreads();
  for (int s = 128; s > 0; s >>= 1) {
    if (tid < s) red[tid] = fmaxf(red[tid], red[tid + s]);
    __syncthreads();
  }
  mx = red[0];
  __syncthreads();
  float sum = 0.0f;
  for (int m = tid; m < kM; m += 256) sum += expf(base[(size_t)m * kFE] - mx);
  red[tid] = sum;
  __syncthreads();
  for (int s = 128; s > 0; s >>= 1) {
    if (tid < s) red[tid] += red[tid + s];
    __syncthreads();
  }
  const float inv = 1.0f / red[0];
  for (int m = tid; m < kM; m += 256)
    obase[(size_t)m * kFE] = expf(base[(size_t)m * kFE] - mx) * inv;
}

// ---------------------------------------------------------------------------
// Input flattening: canonical JAX pytree order (sorted dict keys, {"W","b"}
// per layer with 'W' < 'b'):
//   R, Ra_data, S, node_data, then params: b1, b2, ee1..ee4, ve1, ve2,
//   vv1, vv2, w1_1, w1_2, w2_1, w2_2.
// ---------------------------------------------------------------------------
enum {
  IN_R = 0, IN_RA = 1, IN_S = 2, IN_NODE = 3,
  IN_B1 = 4, IN_B2 = 5,
  IN_EE1_0W = 6,  IN_EE1_0B, IN_EE1_1W, IN_EE1_1B,
  IN_EE2_0W = 10, IN_EE2_0B, IN_EE2_1W, IN_EE2_1B,
  IN_EE3_0W = 14, IN_EE3_0B, IN_EE3_1W, IN_EE3_1B,
  IN_EE4_0W = 18, IN_EE4_0B, IN_EE4_1W, IN_EE4_1B,
  IN_VE1_0W = 22, IN_VE1_0B, IN_VE1_1W, IN_VE1_1B, IN_VE1_2W, IN_VE1_2B,
  IN_VE2_0W = 28, IN_VE2_0B, IN_VE2_1W, IN_VE2_1B, IN_VE2_2W, IN_VE2_2B,
  IN_VV1_0W = 34, IN_VV1_0B, IN_VV1_1W, IN_VV1_1B,
  IN_VV2_0W = 38, IN_VV2_0B, IN_VV2_1W, IN_VV2_1B,
  IN_W1_1 = 42, IN_W1_2 = 43, IN_W2_1 = 44, IN_W2_2 = 45,
};

extern "C" void kernel_launch(void* const* d_in, const int* in_sizes, int n_in,
                              void* d_out, int out_size, void* d_ws, size_t ws_size,
                              hipStream_t stream) {
  (void)in_sizes; (void)n_in; (void)out_size; (void)ws_size;
  auto F = [&](int i) { return (const float*)d_in[i]; };

  // ---- workspace carve-up
  char* ws = (char*)d_ws;
  int* recv = (int*)ws;                 ws += (size_t)kM * sizeof(int);
  int* send = (int*)ws;                 ws += (size_t)kM * sizeof(int);
  float* Hev  = (float*)ws;             ws += (size_t)kB * kN * kQ * sizeof(float);
  float* He2  = (float*)ws;             ws += (size_t)kB * kM * kINTER * sizeof(float);
  float* Hv2  = (float*)ws;             ws += (size_t)kB * kN * kINTER * sizeof(float);
  float* Hev2 = (float*)ws;             ws += (size_t)kB * kN * kQ * sizeof(float);

  // ---- outputs (tuple concat: Hv3, He3, He_logits3)
  float* out    = (float*)d_out;
  float* Hv3    = out;
  float* He3    = out + (size_t)kB * kN * kFV;
  float* logits = He3 + (size_t)kB * kM * kFE;

  hipMemsetAsync(Hev, 0, (size_t)kB * kN * kQ * sizeof(float), stream);
  hipMemsetAsync(Hev2, 0, (size_t)kB * kN * kQ * sizeof(float), stream);

  k_extract_idx<<<kM / 256, 256, 0, stream>>>(F(IN_R), F(IN_S), recv, send);

  S1P p1;
  p1.node = F(IN_NODE); p1.Ra = F(IN_RA); p1.recv = recv; p1.send = send;
  p1.ve10W = F(IN_VE1_0W); p1.ve10b = F(IN_VE1_0B);
  p1.ve11W = F(IN_VE1_1W); p1.ve11b = F(IN_VE1_1B);
  p1.ve12W = F(IN_VE1_2W); p1.ve12b = F(IN_VE1_2B);
  p1.w11 = F(IN_W1_1); p1.w12 = F(IN_W1_2); p1.b1 = F(IN_B1);
  p1.ee10W = F(IN_EE1_0W); p1.ee10b = F(IN_EE1_0B);
  p1.ee11W = F(IN_EE1_1W); p1.ee11b = F(IN_EE1_1B);
  p1.ee20W = F(IN_EE2_0W); p1.ee20b = F(IN_EE2_0B);
  p1.ee21W = F(IN_EE2_1W); p1.ee21b = F(IN_EE2_1B);
  p1.Hev = Hev; p1.He2 = He2;
  k_edge_stage1<<<(kB * kM) / (16 * WPB), TPB, 0, stream>>>(p1);

  k_node_mlp<<<(kB * kN) / (16 * WPB), TPB, 0, stream>>>(
      Hev, F(IN_NODE), F(IN_VV1_0W), F(IN_VV1_0B), F(IN_VV1_1W), F(IN_VV1_1B),
      Hv2, kINTER);

  S2P p2;
  p2.Hv2 = Hv2; p2.He2 = He2; p2.Ra = F(IN_RA); p2.recv = recv; p2.send = send;
  p2.ve20W = F(IN_VE2_0W); p2.ve20b = F(IN_VE2_0B);
  p2.ve21W = F(IN_VE2_1W); p2.ve21b = F(IN_VE2_1B);
  p2.ve22W = F(IN_VE2_2W); p2.ve22b = F(IN_VE2_2B);
  p2.w21 = F(IN_W2_1); p2.w22 = F(IN_W2_2); p2.b2 = F(IN_B2);
  p2.ee30W = F(IN_EE3_0W); p2.ee30b = F(IN_EE3_0B);
  p2.ee31W = F(IN_EE3_1W); p2.ee31b = F(IN_EE3_1B);
  p2.ee40W = F(IN_EE4_0W); p2.ee40b = F(IN_EE4_0B);
  p2.ee41W = F(IN_EE4_1W); p2.ee41b = F(IN_EE4_1B);
  p2.Hev2 = Hev2; p2.logits = logits;
  k_edge_stage2<<<(kB * kM) / (16 * WPB), TPB, 0, stream>>>(p2);

  k_node_mlp<<<(kB * kN) / (16 * WPB), TPB, 0, stream>>>(
      Hev2, F(IN_NODE), F(IN_VV2_0W), F(IN_VV2_0B), F(IN_VV2_1W), F(IN_VV2_1B),
      Hv3, kFV);

  k_softmax_edges<<<kB * kFE, 256, 0, stream>>>(logits, He3);
}